// ConvCaps_semantic_20650202759457
// MI455X (gfx1250) — compile-verified
//
#include <hip/hip_runtime.h>
#include <math.h>

// ---------------------------------------------------------------------------
// Matrix-capsule EM routing (ConvCaps 1x1) for gfx1250.
//  Phase 1: v[b,B,C,4,4] = p_in[b,B] @ W[B,C] via V_WMMA_F32_16X16X4_F32
//           (K=4 exactly matches the pose contraction). Each wave handles 4
//           batch elements; 16 WMMAs/wave produce all 8x8x16 v-values, which
//           are scattered to LDS to convert from WMMA lane layout to a
//           per-(batch,C) thread layout.
//  Phase 2: EM routing, one thread per (batch,C). v[8][16] lives in VGPRs;
//           C-axis reductions use wave32 xor-shuffle butterflies over the
//           aligned 8-lane C-groups. B-axis reductions are in-thread.
//           Transcendentals use hardware v_log_f32/v_exp_f32/v_rcp_f32.
// ---------------------------------------------------------------------------

typedef float v2f __attribute__((ext_vector_type(2)));
typedef float v8f __attribute__((ext_vector_type(8)));

#define NB      8        // B_CAPS
#define NC      8        // C_CAPS
#define PSZ     16       // P*P
#define NITER   3
#define EPSV    1e-8f
#define LAMB    1e-3f
#define LN2PI   1.8378770664093453f
#define XSTR    136      // B_CAPS*(PSIZE+1)
#define BPB     16       // batch elements per block (4 waves * 4)
#define VSTR    1028     // per-batch LDS stride in dwords: 8*8*16 + 4 pad

__device__ __forceinline__ float sum_c(float v) {
    v += __shfl_xor(v, 1, 32);
    v += __shfl_xor(v, 2, 32);
    v += __shfl_xor(v, 4, 32);
    return v;
}
__device__ __forceinline__ float max_c(float v) {
    v = fmaxf(v, __shfl_xor(v, 1, 32));
    v = fmaxf(v, __shfl_xor(v, 2, 32));
    v = fmaxf(v, __shfl_xor(v, 4, 32));
    return v;
}
__device__ __forceinline__ float frcp(float x) {
    return __builtin_amdgcn_rcpf(x);      // v_rcp_f32
}

__global__ __launch_bounds__(128) void caps_em_kernel(
    const float* __restrict__ x, const float* __restrict__ W,
    const float* __restrict__ beta_u, const float* __restrict__ beta_a,
    float* __restrict__ out, int batch)
{
    __shared__ __align__(16) float v_lds[BPB * VSTR];

    const int tid   = threadIdx.x;
    const int wave  = tid >> 5;
    const int lane  = tid & 31;
    const int blk0  = blockIdx.x * BPB;

    // ------------------------ Phase 1: WMMA for v -------------------------
    {
        const int half = lane >> 4;       // selects K pair {0,1} vs {2,3}
        const int M    = lane & 15;       // A row = 4*b_local + i ; also D/B col index
        const int bl4  = M >> 2;
        const int i    = M & 3;
        const int Cl   = M >> 2;          // B/D col: 4*C_local + j
        const int j    = M & 3;
        const int k0   = half * 2;

        int bidx = blk0 + wave * 4 + bl4;
        if (bidx >= batch) bidx = batch - 1;   // clamp (tail blocks)
        const size_t xbase = (size_t)bidx * XSTR;

        #pragma unroll
        for (int Bc = 0; Bc < NB; ++Bc) {
            // A: p_in[bidx, Bc, i, k0..k0+1]  (contiguous pair, 8B aligned)
            const float* xp = x + xbase + Bc * 16 + i * 4 + k0;
            v2f A;
            A.x = xp[0];
            A.y = xp[1];
            #pragma unroll
            for (int h = 0; h < 2; ++h) {
                const int C = 4 * h + Cl;
                // B: W[Bc, C, k, j] at N=(C_local,j), K rows selected by lane half
                const float* wp = W + Bc * (NC * PSZ) + C * PSZ + j;
                v2f Bm;
                Bm.x = wp[k0 * 4];
                Bm.y = wp[(k0 + 1) * 4];
                v8f d = {};
                d = __builtin_amdgcn_wmma_f32_16x16x4_f32(
                        false, A, false, Bm, (short)0, d, false, false);
                // scatter D (8 VGPRs) to LDS in (batch_local, B, C, e) layout
                #pragma unroll
                for (int r = 0; r < 8; ++r) {
                    const int Md = r + 8 * half;        // D row
                    const int bl = wave * 4 + (Md >> 2);
                    const int e  = (Md & 3) * 4 + j;    // i*4 + j
                    v_lds[bl * VSTR + (Bc * NC + C) * PSZ + e] = d[r];
                }
            }
        }
    }
    __syncthreads();

    // ------------------------ Phase 2: EM routing -------------------------
    const int bl = tid >> 3;        // batch_local 0..15
    const int C  = tid & 7;         // this thread's C capsule
    const int b  = blk0 + bl;
    const bool live = (b < batch);
    const size_t xb = (size_t)(live ? b : 0) * XSTR;

    // pull v[B][e] into VGPRs (ds_load_b128 x4 per B)
    float vv[NB][PSZ];
    #pragma unroll
    for (int B = 0; B < NB; ++B) {
        const float4* q = (const float4*)&v_lds[bl * VSTR + (B * NC + C) * PSZ];
        #pragma unroll
        for (int t = 0; t < 4; ++t) {
            float4 f = q[t];
            vv[B][t * 4 + 0] = f.x; vv[B][t * 4 + 1] = f.y;
            vv[B][t * 4 + 2] = f.z; vv[B][t * 4 + 3] = f.w;
        }
    }

    float a_in[NB];
    #pragma unroll
    for (int B = 0; B < NB; ++B) a_in[B] = x[xb + NB * PSZ + B];

    const float bu = beta_u[C];
    const float ba = beta_a[C];

    float r[NB];
    #pragma unroll
    for (int B = 0; B < NB; ++B) r[B] = 1.0f / (float)NC;

    float mu[PSZ], sig[PSZ];
    float a_out = 0.0f;

    #pragma unroll
    for (int it = 0; it < NITER; ++it) {
        // rr = (r*a_in) normalized over C ; r_sum = sum_B rr
        float coeff[NB];
        float r_sum = 0.0f;
        #pragma unroll
        for (int B = 0; B < NB; ++B) {
            float t = r[B] * a_in[B];
            float s = sum_c(t);
            t = t * frcp(s + EPSV);
            coeff[B] = t;            // temporarily rr
            r_sum += t;
        }
        const float inv_rs = frcp(r_sum + EPSV);
        #pragma unroll
        for (int B = 0; B < NB; ++B) coeff[B] *= inv_rs;

        // mu / sigma^2 over B; lgsum = sum_e log(sigma^2)
        float lgsum = 0.0f;
        #pragma unroll
        for (int e = 0; e < PSZ; ++e) {
            float m = 0.0f;
            #pragma unroll
            for (int B = 0; B < NB; ++B) m = fmaf(coeff[B], vv[B][e], m);
            mu[e] = m;
            float s2 = 0.0f;
            #pragma unroll
            for (int B = 0; B < NB; ++B) {
                float d = vv[B][e] - m;
                s2 = fmaf(coeff[B], d * d, s2);
            }
            sig[e] = s2 + EPSV;
            lgsum += __logf(sig[e]);          // v_log_f32
        }

        // activation: cost = r_sum * (16*bu + 0.5*lgsum)
        const float cost = r_sum * fmaf(0.5f, lgsum, 16.0f * bu);
        a_out = frcp(1.0f + __expf(-(LAMB * (ba - cost))));

        if (it < NITER - 1) {
            const float ln_a = __logf(a_out);
            // B-invariant part of sum_e ln_p: -0.5*lgsum - 8*ln(2pi)
            const float base = fmaf(-0.5f, lgsum, -8.0f * LN2PI) + ln_a;
            float n2s[PSZ];
            #pragma unroll
            for (int e = 0; e < PSZ; ++e) n2s[e] = -0.5f * frcp(sig[e]);
            #pragma unroll
            for (int B = 0; B < NB; ++B) {
                float lp = 0.0f;
                #pragma unroll
                for (int e = 0; e < PSZ; ++e) {
                    float d = vv[B][e] - mu[e];
                    lp = fmaf(d * d, n2s[e], lp);   // -(d^2)/(2*sig)
                }
                float la = lp + base;
                float mx = max_c(la);        // softmax over C (8-lane group)
                float ex = __expf(la - mx);  // v_exp_f32
                float se = sum_c(ex);
                r[B] = ex * frcp(se);
            }
        }
    }

    // ------------------------ Outputs (4 tensors, concat) -----------------
    if (live) {
        const size_t N = (size_t)batch;
        float* p_out_p = out;                   // (N, 8, 16)
        float* a_out_p = out + N * 128;         // (N, 8, 1)
        float* out_p   = out + N * 136;         // (N, 8, 17)
        float* out4    = out + N * 272;         // (N, 136)

        // 16B-aligned tensors: float4 stores (global_store_b128)
        float4* pp = (float4*)(p_out_p + (size_t)b * 128 + C * 16);
        float4* o4 = (float4*)(out4   + (size_t)b * 136 + C * 16);
        #pragma unroll
        for (int t = 0; t < 4; ++t) {
            float4 m4;
            m4.x = mu[t * 4 + 0]; m4.y = mu[t * 4 + 1];
            m4.z = mu[t * 4 + 2]; m4.w = mu[t * 4 + 3];
            pp[t] = m4;
            o4[t] = m4;
        }
        // 17-stride tensor: scalar stores (unaligned for b128)
        #pragma unroll
        for (int e = 0; e < PSZ; ++e)
            out_p[(size_t)b * 136 + C * 17 + e] = mu[e];

        a_out_p[(size_t)b * 8 + C]             = a_out;
        out_p  [(size_t)b * 136 + C * 17 + 16] = a_out;
        out4   [(size_t)b * 136 + 128 + C]     = a_out;
    }
}

extern "C" void kernel_launch(void* const* d_in, const int* in_sizes, int n_in,
                              void* d_out, int out_size, void* d_ws, size_t ws_size,
                              hipStream_t stream) {
    const float* x  = (const float*)d_in[0];
    const float* W  = (const float*)d_in[1];
    const float* bu = (const float*)d_in[2];
    const float* ba = (const float*)d_in[3];
    const int batch = in_sizes[0] / XSTR;
    const int blocks = (batch + BPB - 1) / BPB;
    caps_em_kernel<<<blocks, 128, 0, stream>>>(x, W, bu, ba, (float*)d_out, batch);
}